// RecurrentCrossFullAttention_24730421690663
// MI455X (gfx1250) — compile-verified
//
#include <hip/hip_runtime.h>
#include <math.h>

typedef __attribute__((ext_vector_type(2))) float v2f;
typedef __attribute__((ext_vector_type(8))) float v8f;

#define N_ 32
#define S_ 4096
#define H_ 16
#define E_ 64
#define D_ 64
#define SPLITS 16
#define CHUNK (S_ / SPLITS)   /* 256 s-positions per block      */
#define TS 16                 /* s-tile per WMMA group          */
#define TILES (CHUNK / TS)    /* 16 tiles per block             */
#define WSTRIDE 68            /* 64 O + m + l + pad, per partial */

// ---------------------------------------------------------------------------
// Kernel 1: per (split, batch) block; wave w handles head h=w over 256 keys.
// Online-softmax flash decoding; QK and AV tiles done with f32 WMMA 16x16x4.
// ---------------------------------------------------------------------------
__launch_bounds__(512, 1)
__global__ void attn_partial(const float* __restrict__ q_,
                             const float* __restrict__ k_,
                             const float* __restrict__ v_,
                             const int*   __restrict__ len_,
                             float*       __restrict__ ws) {
  const int split = blockIdx.x;
  const int n     = blockIdx.y;
  const int w     = threadIdx.x >> 5;   // wave id == head id
  const int lane  = threadIdx.x & 31;
  const int half  = lane >> 4;          // 0: lanes 0-15, 1: lanes 16-31
  const int col   = lane & 15;
  const int h     = w;
  const int s0    = split * CHUNK;
  const int len   = len_[n];

  const float temp = 0.125f;            // 1/sqrt(E=64)

  const float* q  = q_ + ((size_t)n * H_ + h) * E_;
  const float* kb = k_ + (((size_t)n * S_ + s0) * H_ + h) * E_;
  const float* vb = v_ + (((size_t)n * S_ + s0) * H_ + h) * D_;
  const size_t rs = (size_t)H_ * E_;    // 1024 floats between s rows

  __shared__ float pshare[16][16];      // per-wave softmax-weight exchange

  // B operand for QK (q pre-scaled by temp), laid out per ISA f32 B layout:
  // lane holds B[2*half + vgpr, col] = q[4*st + 2*half + vgpr] (col-replicated)
  v2f bq[16];
#pragma unroll
  for (int st = 0; st < 16; ++st) {
    const int e0 = 4 * st + 2 * half;
    bq[st].x = q[e0]     * temp;
    bq[st].y = q[e0 + 1] * temp;
  }

  v8f o0 = {}, o1 = {}, o2 = {}, o3 = {};   // O accumulators, 4 x 16 d-chunk
  float m_run = -1e30f, l_run = 0.0f;

  for (int t = 0; t < TILES; ++t) {
    const int st0 = t * TS;

    // ---- QK: D[s,*] = sum_e K[s,e] * q[e], 16 chained K=4 WMMAs ----------
    // A layout: lane L holds A[L%16, 2*half + vgpr] = K[s_tile + col, e]
    const float* krow = kb + (size_t)(st0 + col) * rs + 2 * half;
    if (t + 1 < TILES) {                 // stream hint for next tile
      __builtin_prefetch(kb + (size_t)(st0 + TS + col) * rs, 0, 1);
      __builtin_prefetch(vb + (size_t)(st0 + TS + col) * rs, 0, 1);
    }
    v8f c = {};
#pragma unroll
    for (int st = 0; st < 16; ++st) {
      v2f a = *(const v2f*)(krow + 4 * st);
      c = __builtin_amdgcn_wmma_f32_16x16x4_f32(
              false, a, false, bq[st], (short)0, c, false, false);
    }

    // ---- mask + online softmax update --------------------------------
    // lane holds rows m = v + 8*half (columns replicated)
    float p[8];
    float tmax = -1e30f;
#pragma unroll
    for (int v = 0; v < 8; ++v) {
      const int sabs = s0 + st0 + v + 8 * half;
      const float sc = (sabs < len) ? c[v] : -1e30f;
      p[v] = sc;
      tmax = fmaxf(tmax, sc);
    }
    tmax = fmaxf(tmax, __shfl_xor(tmax, 16, 32));
    const float m_new = fmaxf(m_run, tmax);
    const float scale = __expf(m_run - m_new);
    float psum = 0.0f;
#pragma unroll
    for (int v = 0; v < 8; ++v) {
      const int sabs = s0 + st0 + v + 8 * half;
      const float pv = (sabs < len) ? __expf(p[v] - m_new) : 0.0f;
      p[v] = pv;
      psum += pv;
    }
    psum += __shfl_xor(psum, 16, 32);
    l_run = l_run * scale + psum;
    m_run = m_new;

    // ---- redistribute p into B lane-layout via per-wave LDS ----------
#pragma unroll
    for (int v = 0; v < 8; ++v) pshare[w][v + 8 * half] = p[v];
    __builtin_amdgcn_wave_barrier();     // LDS is in-order within a wave
    v2f bp[4];
#pragma unroll
    for (int kg = 0; kg < 4; ++kg)
      bp[kg] = *(const v2f*)&pshare[w][kg * 4 + 2 * half];
    __builtin_amdgcn_wave_barrier();

    // ---- rescale running output by exp(m_old - m_new) ----------------
#pragma unroll
    for (int v = 0; v < 8; ++v) {
      o0[v] *= scale; o1[v] *= scale; o2[v] *= scale; o3[v] *= scale;
    }

    // ---- AV: O[d] += sum_s p[s] * V[s,d], A = V-tile, B = p ----------
    // A layout: lane holds A[col, 2*half+vgpr] = V[s_tile + k, h, dc*16+col]
#pragma unroll
    for (int kg = 0; kg < 4; ++kg) {
      const float* r0 = vb + (size_t)(st0 + kg * 4 + 2 * half) * rs + col;
      const float* r1 = r0 + rs;
      v2f a;
      a.x = r0[0];  a.y = r1[0];
      o0 = __builtin_amdgcn_wmma_f32_16x16x4_f32(
               false, a, false, bp[kg], (short)0, o0, false, false);
      a.x = r0[16]; a.y = r1[16];
      o1 = __builtin_amdgcn_wmma_f32_16x16x4_f32(
               false, a, false, bp[kg], (short)0, o1, false, false);
      a.x = r0[32]; a.y = r1[32];
      o2 = __builtin_amdgcn_wmma_f32_16x16x4_f32(
               false, a, false, bp[kg], (short)0, o2, false, false);
      a.x = r0[48]; a.y = r1[48];
      o3 = __builtin_amdgcn_wmma_f32_16x16x4_f32(
               false, a, false, bp[kg], (short)0, o3, false, false);
    }
  }

  // ---- write partial (O[64], m, l) for this (n, h, split) --------------
  // D layout: lane holds rows d = v + 8*half (columns identical)
  float* wsp = ws + (((size_t)n * H_ + h) * SPLITS + split) * WSTRIDE;
  if (col == 0) {
#pragma unroll
    for (int v = 0; v < 8; ++v) {
      const int r = v + 8 * half;
      wsp[ 0 + r] = o0[v];
      wsp[16 + r] = o1[v];
      wsp[32 + r] = o2[v];
      wsp[48 + r] = o3[v];
    }
    if (half == 0) { wsp[64] = m_run; wsp[65] = l_run; }
  }
}

// ---------------------------------------------------------------------------
// Kernel 2: merge the SPLITS partials per (n,h) with log-sum-exp weighting.
// ---------------------------------------------------------------------------
__global__ void attn_reduce(const float* __restrict__ ws,
                            float*       __restrict__ out) {
  const int nh = blockIdx.x;            // n*H + h
  const int d  = threadIdx.x;           // 0..63
  const float* base = ws + (size_t)nh * SPLITS * WSTRIDE;

  float M = -1e30f;
  for (int i = 0; i < SPLITS; ++i) M = fmaxf(M, base[i * WSTRIDE + 64]);
  float acc = 0.0f, l = 0.0f;
  for (int i = 0; i < SPLITS; ++i) {
    const float wgt = __expf(base[i * WSTRIDE + 64] - M);
    l   += base[i * WSTRIDE + 65] * wgt;
    acc += base[i * WSTRIDE + d]  * wgt;
  }
  out[(size_t)nh * D_ + d] = acc / l;
}

// ---------------------------------------------------------------------------
extern "C" void kernel_launch(void* const* d_in, const int* in_sizes, int n_in,
                              void* d_out, int out_size, void* d_ws, size_t ws_size,
                              hipStream_t stream) {
  const float* query  = (const float*)d_in[0];   // [N,H,E]
  const float* keys   = (const float*)d_in[1];   // [N,S,H,E]
  const float* values = (const float*)d_in[2];   // [N,S,H,D]
  const int*   klen   = (const int*)d_in[3];     // [N]
  float* out = (float*)d_out;                    // [N,H,D]
  float* ws  = (float*)d_ws;                     // >= N*H*SPLITS*WSTRIDE*4 B

  attn_partial<<<dim3(SPLITS, N_), 512, 0, stream>>>(query, keys, values, klen, ws);
  attn_reduce<<<N_ * H_, D_, 0, stream>>>(ws, out);
}